// ROIPooling_44006234915658
// MI455X (gfx1250) — compile-verified
//
#include <hip/hip_runtime.h>
#include <stdint.h>

#define POOL 7
#define FW   128
#define FH   128
#define FC   512

typedef unsigned int u32x4 __attribute__((ext_vector_type(4)));
typedef int          i32x8 __attribute__((ext_vector_type(8)));
typedef int          i32x4 __attribute__((ext_vector_type(4)));
typedef float        fvec4 __attribute__((ext_vector_type(4)));

__device__ __forceinline__ unsigned rfl(unsigned v) {
  return __builtin_amdgcn_readfirstlane(v);
}

// Matches reference _half_pixel_coords: c = (j+0.5)*size/POOL - 0.5, clipped, + start
__device__ __forceinline__ float coord_at(int j, int start, int size) {
  float sf = (float)size;
  float c  = ((float)j + 0.5f) * (sf * (1.0f / (float)POOL)) - 0.5f;
  c = fminf(fmaxf(c, 0.0f), sf - 1.0f);
  return c + (float)start;
}

__device__ __forceinline__ float lerpf(float a, float b, float w) {
  return a + (b - a) * w;  // exact reference formulation
}

__device__ __forceinline__ void tdm_gather(u32x4 g0, i32x8 g1, i32x4 g2, i32x4 g3) {
#if __clang_major__ >= 23
  i32x8 gx = {0, 0, 0, 0, 0, 0, 0, 0};
  __builtin_amdgcn_tensor_load_to_lds(g0, g1, g2, g3, gx, 0);
#else
  __builtin_amdgcn_tensor_load_to_lds(g0, g1, g2, g3, 0);
#endif
}

// One block = one (roi, py) output row: stages 28 tap columns (4 taps x 7 px,
// 512 f32 each = 56 KB) into LDS via two TDM gather ops (issued by wave 0 and
// wave 1 in parallel, each waiting on its own TENSORcnt), then 8 waves compute
// 7 x 512 bilinear outputs with ds_load_b128 + NT global_store_b128.
__global__ __launch_bounds__(256) void roialign_tdm_kernel(
    const float* __restrict__ fm, const int* __restrict__ rois,
    float* __restrict__ out) {
  __shared__ fvec4 tile[28 * (FC / 4)];  // column k at tile[k*128 .. k*128+127]

  const int blk = blockIdx.x;
  const int roi = blk / POOL;
  const int py  = blk % POOL;

  // ROI (uniform per block -> scalar loads)
  const int rx1 = rois[roi * 4 + 0];
  const int ry1 = rois[roi * 4 + 1];
  const int rx2 = rois[roi * 4 + 2];
  const int ry2 = rois[roi * 4 + 3];

  // y taps (uniform for the whole block)
  const float ysf = coord_at(py, ry1, ry2 - ry1);
  const int   y0  = (int)floorf(ysf);
  const int   y1i = min(y0 + 1, ry2 - 1);
  const float wy  = ysf - (float)y0;

  const int wave = (int)(threadIdx.x >> 5);
  if (wave < 2) {  // waves 0 and 1 each issue one TDM gather
    // 28 gather row indices packed as 16-bit pairs: 4 taps per px
    unsigned pk[14];
#pragma unroll
    for (int px = 0; px < POOL; ++px) {
      float xsf = coord_at(px, rx1, rx2 - rx1);
      int   x0  = (int)floorf(xsf);
      int   x1  = min(x0 + 1, rx2 - 1);
      unsigned tl = (unsigned)(y0  * FW + x0);
      unsigned tr = (unsigned)(y0  * FW + x1);
      unsigned bl = (unsigned)(y1i * FW + x0);
      unsigned br = (unsigned)(y1i * FW + x1);
      pk[px * 2 + 0] = tl | (tr << 16);
      pk[px * 2 + 1] = bl | (br << 16);
    }

    const unsigned ldsBase = rfl((unsigned)(uintptr_t)&tile[0]);  // flat low32 == LDS offset
    const uint64_t ga      = (uint64_t)(uintptr_t)fm;

    // ---- D# group1: gather mode, 4B elements, 512-wide rows, 16384 rows ----
    i32x8 g1;
    g1[0] = 0x00020000;                        // data_size=2 (4 bytes); wg_mask=0
    g1[1] = (int)((FC & 0xFFFF) << 16);        // tensor_dim0[15:0] = 512
    g1[2] = (int)(((FH * FW) & 0xFFFF) << 16); // tensor_dim1[15:0] = 16384
    g1[3] = (int)((FC & 0xFFFF) << 16);        // tile_dim0 = 512
    g1[4] = (wave == 0) ? 16 : 12;             // tile_dim1 = #valid indices
    g1[5] = FC;                                // tensor_dim0_stride[31:0] = 512
    g1[6] = 0;
    g1[7] = 0;

    // ---- D# group0: count=1, gather_mode=1, 16-bit indices, type=2 ----
    u32x4 g0;
    g0[0] = 0x80000001u;
    g0[1] = (wave == 0) ? ldsBase : (ldsBase + 16u * FC * 4u);
    g0[2] = rfl((unsigned)(ga & 0xFFFFFFFFu));
    g0[3] = rfl((unsigned)((ga >> 32) & 0x01FFFFFFu) | 0x80000000u);

    // ---- indices: wave 0 -> columns 0..15 (pk[0..7]);
    //               wave 1 -> columns 16..27 (pk[8..13], pad 0) ----
    i32x4 g2, g3;
    if (wave == 0) {
#pragma unroll
      for (int i = 0; i < 4; ++i) {
        g2[i] = (int)rfl(pk[i]);
        g3[i] = (int)rfl(pk[4 + i]);
      }
    } else {
#pragma unroll
      for (int i = 0; i < 4; ++i) {
        g2[i] = (int)rfl(pk[8 + i]);
        g3[i] = (int)rfl((12 + i < 14) ? pk[12 + i] : 0u);
      }
    }
    tdm_gather(g0, g1, g2, g3);

    __builtin_amdgcn_s_wait_tensorcnt(0);  // each wave waits only for its own DMA
  }
  __syncthreads();  // release compute waves once the LDS tile is ready

  // ---- compute: 7 px * 128 float4 = 896 vec4 outputs per block ----
  fvec4* o4 = (fvec4*)out;
  const size_t obase = (size_t)blk * (POOL * (FC / 4));
  for (int i = threadIdx.x; i < POOL * (FC / 4); i += 256) {
    const int px = i >> 7;  // i / 128
    const int c4 = i & 127;

    const float xsf = coord_at(px, rx1, rx2 - rx1);
    const float wx  = xsf - floorf(xsf);

    const int cb = (px * 4) * (FC / 4) + c4;
    fvec4 tl = tile[cb + 0 * (FC / 4)];
    fvec4 tr = tile[cb + 1 * (FC / 4)];
    fvec4 bl = tile[cb + 2 * (FC / 4)];
    fvec4 br = tile[cb + 3 * (FC / 4)];

    fvec4 r;
    r.x = lerpf(lerpf(tl.x, tr.x, wx), lerpf(bl.x, br.x, wx), wy);
    r.y = lerpf(lerpf(tl.y, tr.y, wx), lerpf(bl.y, br.y, wx), wy);
    r.z = lerpf(lerpf(tl.z, tr.z, wx), lerpf(bl.z, br.z, wx), wy);
    r.w = lerpf(lerpf(tl.w, tr.w, wx), lerpf(bl.w, br.w, wx), wy);

    // Non-temporal: stream the 103 MB output past L2 so the 32 MB feature
    // map stays resident (192 MB global L2).
    __builtin_nontemporal_store(r, &o4[obase + (size_t)px * (FC / 4) + c4]);
  }
}

extern "C" void kernel_launch(void* const* d_in, const int* in_sizes, int n_in,
                              void* d_out, int out_size, void* d_ws, size_t ws_size,
                              hipStream_t stream) {
  const float* fm   = (const float*)d_in[0];
  const int*   rois = (const int*)d_in[1];
  float*       out  = (float*)d_out;
  const int nrois = in_sizes[1] / 4;  // (N,4) int32
  dim3 grid((unsigned)(nrois * POOL));
  dim3 block(256);
  hipLaunchKernelGGL(roialign_tdm_kernel, grid, block, 0, stream, fm, rois, out);
}